// GraphSAGE_46205258170447
// MI455X (gfx1250) — compile-verified
//
#include <hip/hip_runtime.h>
#include <hip/hip_bf16.h>

#define N_NODES 50000
#define D0 64
#define D1 128
#define D2 40

typedef __attribute__((ext_vector_type(2))) float v2f;
typedef __attribute__((ext_vector_type(8))) float v8f;

#define WMMA_F32(A, B, C) \
  __builtin_amdgcn_wmma_f32_16x16x4_f32(false, (A), false, (B), (short)0, (C), false, false)

// ---------------- utility kernels ----------------

__global__ void zero_f32(float* __restrict__ p, long n) {
  long i = (long)blockIdx.x * blockDim.x + threadIdx.x;
  long stride = (long)gridDim.x * blockDim.x;
  for (; i < n; i += stride) p[i] = 0.0f;
}

__global__ void deg_count(const int* __restrict__ dst, float* __restrict__ deg, int E) {
  int i = blockIdx.x * blockDim.x + threadIdx.x;
  if (i < E) atomicAdd(&deg[dst[i]], 1.0f);
}

__global__ void inv_deg(float* __restrict__ deg, int n) {
  int i = blockIdx.x * blockDim.x + threadIdx.x;
  if (i < n) deg[i] = 1.0f / fmaxf(deg[i], 1.0f);
}

// gather x[src] and scatter-add into sum[dst]; D4 = feature dim / 4, each
// thread moves one float4 chunk of one edge via global_atomic_add_f32.
__global__ void scatter_sum(const float* __restrict__ x, const int* __restrict__ src,
                            const int* __restrict__ dst, float* __restrict__ sum,
                            int E, int D4) {
  long idx = (long)blockIdx.x * blockDim.x + threadIdx.x;
  long total = (long)E * D4;
  if (idx >= total) return;
  int e = (int)(idx / D4);
  int c = (int)(idx % D4);
  int s = src[e], d = dst[e];
  float4 v = ((const float4*)x)[(long)s * D4 + c];
  float* out = sum + ((long)d * D4 + c) * 4;
  atomicAdd(out + 0, v.x);
  atomicAdd(out + 1, v.y);
  atomicAdd(out + 2, v.z);
  atomicAdd(out + 3, v.w);
}

// ---------------- WMMA fp32 GEMM kernels ----------------
// V_WMMA_F32_16X16X4_F32 fragments (wave32):
//   A (16x4): lanes 0-15 hold {K=k,k+1} of row lane&15; lanes 16-31 {K=k+2,k+3}
//   B (4x16): VGPR0 = row (k + half*2), VGPR1 = row (k + half*2 + 1), col = lane&15
//   C/D (16x16): VGPR v = row v + half*8, col = lane&15
// Register blocking: one wave computes several 16x16 N-tiles so the A fragment
// is loaded once per K-step and reused across all WMMAs of that step.

// h1[M,128] = relu( (sum1 * inv) @ Wl + b + x @ Wr ), K = 64 each.
// Each wave: one 16-row M tile x four 16-col N tiles (64 cols).
__global__ void __launch_bounds__(32)
sage_gemm1(const float* __restrict__ sum1, const float* __restrict__ inv,
           const float* __restrict__ x,
           const float* __restrict__ Wl, const float* __restrict__ Wr,
           const float* __restrict__ b, float* __restrict__ h1) {
  const int lane = threadIdx.x;
  const int half = lane >> 4;
  const int l15  = lane & 15;
  const int rowA = blockIdx.x * 16 + l15;
  const int colB = blockIdx.y * 64 + l15;       // base column of 4 N-tiles
  const float scale = inv[rowA];
  const float* __restrict__ sRow = sum1 + (long)rowA * D0;
  const float* __restrict__ xRow = x    + (long)rowA * D0;

  v8f ac0 = {}, ac1 = {}, ac2 = {}, ac3 = {};

#pragma unroll
  for (int k0 = 0; k0 < D0; k0 += 4) {          // AGG @ Wl
    const int ka = k0 + half * 2;
    v2f a;
    a.x = sRow[ka] * scale;
    a.y = sRow[ka + 1] * scale;
    const float* w0 = Wl + (long)ka * D1 + colB;
    const float* w1 = w0 + D1;
    v2f b0, b1, b2, b3;
    b0.x = w0[0];  b0.y = w1[0];
    b1.x = w0[16]; b1.y = w1[16];
    b2.x = w0[32]; b2.y = w1[32];
    b3.x = w0[48]; b3.y = w1[48];
    ac0 = WMMA_F32(a, b0, ac0);
    ac1 = WMMA_F32(a, b1, ac1);
    ac2 = WMMA_F32(a, b2, ac2);
    ac3 = WMMA_F32(a, b3, ac3);
  }
#pragma unroll
  for (int k0 = 0; k0 < D0; k0 += 4) {          // X @ Wr
    const int ka = k0 + half * 2;
    v2f a;
    a.x = xRow[ka];
    a.y = xRow[ka + 1];
    const float* w0 = Wr + (long)ka * D1 + colB;
    const float* w1 = w0 + D1;
    v2f b0, b1, b2, b3;
    b0.x = w0[0];  b0.y = w1[0];
    b1.x = w0[16]; b1.y = w1[16];
    b2.x = w0[32]; b2.y = w1[32];
    b3.x = w0[48]; b3.y = w1[48];
    ac0 = WMMA_F32(a, b0, ac0);
    ac1 = WMMA_F32(a, b1, ac1);
    ac2 = WMMA_F32(a, b2, ac2);
    ac3 = WMMA_F32(a, b3, ac3);
  }

#pragma unroll
  for (int v = 0; v < 8; ++v) {
    const long rowD = blockIdx.x * 16 + v + half * 8;
    float* o = h1 + rowD * D1 + colB;
    o[0]  = fmaxf(ac0[v] + b[colB],      0.0f);
    o[16] = fmaxf(ac1[v] + b[colB + 16], 0.0f);
    o[32] = fmaxf(ac2[v] + b[colB + 32], 0.0f);
    o[48] = fmaxf(ac3[v] + b[colB + 48], 0.0f);
  }
}

// h2[M,40] = (sum2 * inv) @ Wl + b + h1 @ Wr, K = 128 each, N padded 40->48.
// Each wave: one 16-row M tile x all three 16-col N-tiles (tile 2 guarded).
__global__ void __launch_bounds__(32)
sage_gemm2(const float* __restrict__ sum2, const float* __restrict__ inv,
           const float* __restrict__ h1,
           const float* __restrict__ Wl, const float* __restrict__ Wr,
           const float* __restrict__ b, float* __restrict__ h2) {
  const int lane = threadIdx.x;
  const int half = lane >> 4;
  const int l15  = lane & 15;
  const int rowA = blockIdx.x * 16 + l15;
  const bool c2ok = (32 + l15) < D2;            // cols 32..39 valid
  const float scale = inv[rowA];
  const float* __restrict__ sRow = sum2 + (long)rowA * D1;
  const float* __restrict__ hRow = h1   + (long)rowA * D1;

  v8f ac0 = {}, ac1 = {}, ac2 = {};

#pragma unroll 8
  for (int k0 = 0; k0 < D1; k0 += 4) {          // AGG2 @ Wl
    const int ka = k0 + half * 2;
    v2f a;
    a.x = sRow[ka] * scale;
    a.y = sRow[ka + 1] * scale;
    const float* w0 = Wl + (long)ka * D2 + l15;
    const float* w1 = w0 + D2;
    v2f b0, b1, b2;
    b0.x = w0[0];  b0.y = w1[0];
    b1.x = w0[16]; b1.y = w1[16];
    b2.x = c2ok ? w0[32] : 0.0f;
    b2.y = c2ok ? w1[32] : 0.0f;
    ac0 = WMMA_F32(a, b0, ac0);
    ac1 = WMMA_F32(a, b1, ac1);
    ac2 = WMMA_F32(a, b2, ac2);
  }
#pragma unroll 8
  for (int k0 = 0; k0 < D1; k0 += 4) {          // H1 @ Wr
    const int ka = k0 + half * 2;
    v2f a;
    a.x = hRow[ka];
    a.y = hRow[ka + 1];
    const float* w0 = Wr + (long)ka * D2 + l15;
    const float* w1 = w0 + D2;
    v2f b0, b1, b2;
    b0.x = w0[0];  b0.y = w1[0];
    b1.x = w0[16]; b1.y = w1[16];
    b2.x = c2ok ? w0[32] : 0.0f;
    b2.y = c2ok ? w1[32] : 0.0f;
    ac0 = WMMA_F32(a, b0, ac0);
    ac1 = WMMA_F32(a, b1, ac1);
    ac2 = WMMA_F32(a, b2, ac2);
  }

  const float bi0 = b[l15];
  const float bi1 = b[16 + l15];
  const float bi2 = c2ok ? b[32 + l15] : 0.0f;
#pragma unroll
  for (int v = 0; v < 8; ++v) {
    const long rowD = blockIdx.x * 16 + v + half * 8;
    float* o = h2 + rowD * D2 + l15;
    o[0]  = ac0[v] + bi0;
    o[16] = ac1[v] + bi1;
    if (c2ok) o[32] = ac2[v] + bi2;
  }
}

// per-node log_softmax over 40 classes
__global__ void logsoftmax40(const float* __restrict__ h, float* __restrict__ out, int n) {
  int i = blockIdx.x * blockDim.x + threadIdx.x;
  if (i >= n) return;
  const float* r = h + (long)i * D2;
  float m = -INFINITY;
#pragma unroll 8
  for (int j = 0; j < D2; ++j) m = fmaxf(m, r[j]);
  float s = 0.0f;
#pragma unroll 8
  for (int j = 0; j < D2; ++j) s += __expf(r[j] - m);
  const float ls = __logf(s);
  float* o = out + (long)i * D2;
#pragma unroll 8
  for (int j = 0; j < D2; ++j) o[j] = r[j] - m - ls;
}

// ---------------- launch ----------------

extern "C" void kernel_launch(void* const* d_in, const int* in_sizes, int n_in,
                              void* d_out, int out_size, void* d_ws, size_t ws_size,
                              hipStream_t stream) {
  const float* x    = (const float*)d_in[0];
  const int*   edge = (const int*)  d_in[1];   // [2, E] flat: src row then dst row
  const float* W1l  = (const float*)d_in[2];
  const float* W1r  = (const float*)d_in[3];
  const float* b1   = (const float*)d_in[4];
  const float* W2l  = (const float*)d_in[5];
  const float* W2r  = (const float*)d_in[6];
  const float* b2   = (const float*)d_in[7];
  const int E = in_sizes[1] / 2;
  const int* src = edge;
  const int* dst = edge + E;

  // workspace layout (floats): [deg | sum1 | sum2 | h1]; first three zeroed
  float* ws   = (float*)d_ws;
  float* deg  = ws;
  float* sum1 = deg  + N_NODES;
  float* sum2 = sum1 + (long)N_NODES * D0;
  float* h1   = sum2 + (long)N_NODES * D1;

  float* h2  = (float*)d_out;                 // first output: h  [N,40]
  float* lsm = h2 + (long)N_NODES * D2;       // second output: log_softmax(h)

  const long zlen = (long)N_NODES * (1 + D0 + D1);
  zero_f32<<<2048, 256, 0, stream>>>(ws, zlen);

  deg_count<<<(E + 255) / 256, 256, 0, stream>>>(dst, deg, E);
  inv_deg<<<(N_NODES + 255) / 256, 256, 0, stream>>>(deg, N_NODES);

  {
    long total = (long)E * (D0 / 4);
    scatter_sum<<<(int)((total + 255) / 256), 256, 0, stream>>>(x, src, dst, sum1, E, D0 / 4);
  }

  dim3 g1(N_NODES / 16, D1 / 64);             // 3125 x 2; each wave does 16x64
  sage_gemm1<<<g1, 32, 0, stream>>>(sum1, deg, x, W1l, W1r, b1, h1);

  {
    long total = (long)E * (D1 / 4);
    scatter_sum<<<(int)((total + 255) / 256), 256, 0, stream>>>(h1, src, dst, sum2, E, D1 / 4);
  }

  sage_gemm2<<<N_NODES / 16, 32, 0, stream>>>(sum2, deg, h1, W2l, W2r, b2, h2);

  logsoftmax40<<<(N_NODES + 255) / 256, 256, 0, stream>>>(h2, lsm, N_NODES);
}